// BlinnPhongShaderEnvMap_46574625358132
// MI455X (gfx1250) — compile-verified
//
#include <hip/hip_runtime.h>
#include <hip/hip_bf16.h>
#include <math.h>

// ---------------------------------------------------------------------------
// Blinn-Phong env-map shading on gfx1250 using V_WMMA_F32_16X16X4_F32.
//
// Per wave32: 16 pixels. Per light-tile of 16:
//   nL = A_n(16x4) x B_L(4x16)   (WMMA, K=3 padded to 4)
//   vL = A_v(16x4) x B_L(4x16)   (WMMA)
// Epilogue identities:
//   n.Hv        = (n.v + n.L) / max(||v+L||, eps)
//   ||v+L||^2   = ||v||^2 + 2 v.L + 1            (lights are unit vectors)
//   1/max(sqrt(x), e) = rsq(max(x, e^2))         -> single v_rsq_f32
//   pow(s, p)   = exp2(p * log2(s))              -> v_log_f32 + v_exp_f32
// ---------------------------------------------------------------------------

typedef float v2f __attribute__((ext_vector_type(2)));
typedef float v8f __attribute__((ext_vector_type(8)));

#define HH 512
#define WW 512
#define NPIX (HH * WW)
#define KL 128
#define LTILES (KL / 16)

__device__ __forceinline__ float fast_rsq_clamped(float x, float eps2) {
    // 1 / max(sqrt(x), sqrt(eps2)) without IEEE div/sqrt expansions
    return __builtin_amdgcn_rsqf(fmaxf(x, eps2));   // v_rsq_f32
}

__global__ __launch_bounds__(256) void blinn_wmma_kernel(
    const float* __restrict__ pn,      // (H,W,3) pixel normals
    const float* __restrict__ pd,      // (H,W,3) pixel directions
    const float* __restrict__ cam,     // (3,)
    const float* __restrict__ ldir,    // (K,3) unit light directions
    const float* __restrict__ lcol,    // (K,3) light colors
    const float* __restrict__ shin_p,  // (1,)
    const float* __restrict__ kd_p,    // (1,)
    const float* __restrict__ ks_p,    // (1,)
    float* __restrict__ out)           // colors (NPIX*3) ++ normals (NPIX*3)
{
    __shared__ float s_ld[KL * 3];
    __shared__ float s_lc[KL * 3];

    const int tid = threadIdx.x;
    for (int i = tid; i < KL * 3; i += 256) {
        s_ld[i] = ldir[i];
        s_lc[i] = lcol[i];
    }
    __syncthreads();

    const int lane     = tid & 31;
    const bool hi      = lane >= 16;         // second 16-lane half
    const int  m       = lane & 15;          // pixel-in-tile this lane mirrors
    const int  wave    = blockIdx.x * 8 + (tid >> 5);
    const int  pixBase = wave * 16;          // grid sized exactly: no tail
    const int  p       = pixBase + m;

    // ---- per-pixel setup (both halves load the same 16 pixels; L2 hit) ----
    float nx = pn[p * 3 + 0], ny = pn[p * 3 + 1], nz = pn[p * 3 + 2];
    float invn = fast_rsq_clamped(nx * nx + ny * ny + nz * nz, 1e-12f);
    nx *= invn; ny *= invn; nz *= invn;

    float vx = cam[0] - pd[p * 3 + 0];
    float vy = cam[1] - pd[p * 3 + 1];
    float vz = cam[2] - pd[p * 3 + 2];
    float invv = fast_rsq_clamped(vx * vx + vy * vy + vz * vz, 1e-12f);
    vx *= invv; vy *= invv; vz *= invv;

    float nv = nx * vx + ny * vy + nz * vz;  // n . v  (per pixel)
    float vv = vx * vx + vy * vy + vz * vz;  // ||v||^2 (~1, kept exact)

    // second output: normalized normals (lanes 0-15 only; exec restored after)
    if (!hi) {
        size_t nb = (size_t)NPIX * 3 + (size_t)p * 3;
        out[nb + 0] = nx;
        out[nb + 1] = ny;
        out[nb + 2] = nz;
    }

    // A-matrix 16x4 f32 layout: lanes 0-15 hold {K=0,K=1}, lanes 16-31 {K=2,K=3}
    v2f An = hi ? (v2f){nz, 0.0f} : (v2f){nx, ny};
    v2f Av = hi ? (v2f){vz, 0.0f} : (v2f){vx, vy};

    // Broadcast per-pixel scalars to the C/D row layout:
    // C/D f32 16x16: lane holds column N=lane%16; VGPR r holds row M=r (+8 for hi)
    const int rbase = hi ? 8 : 0;
    float nvm[8], vvm[8];
#pragma unroll
    for (int r = 0; r < 8; ++r) {
        nvm[r] = __shfl(nv, rbase + r, 32);
        vvm[r] = __shfl(vv, rbase + r, 32);
    }

    const float shin = shin_p[0];
    const float kd   = kd_p[0];
    const float ks   = ks_p[0];
    const float nf   = (shin + 2.0f) / (4.0f * (2.0f - __expf(-shin * 0.5f)));
    const float ksnf = ks * nf;

    float accR[8], accG[8], accB[8];
#pragma unroll
    for (int r = 0; r < 8; ++r) { accR[r] = 0.0f; accG[r] = 0.0f; accB[r] = 0.0f; }

    // ------------------------- light-tile loop -----------------------------
    for (int t = 0; t < LTILES; ++t) {
        const int li = t * 16 + m;  // this lane's light column (global index)
        const float Lx = s_ld[li * 3 + 0];
        const float Ly = s_ld[li * 3 + 1];
        const float Lz = s_ld[li * 3 + 2];

        // B-matrix 4x16 f32: lanes 0-15 hold rows {K=0,K=1}, lanes 16-31 {K=2,K=3}
        v2f B = hi ? (v2f){Lz, 0.0f} : (v2f){Lx, Ly};

        v8f cnl = {};
        v8f cvl = {};
        cnl = __builtin_amdgcn_wmma_f32_16x16x4_f32(
            false, An, false, B, (short)0, cnl, false, false);
        cvl = __builtin_amdgcn_wmma_f32_16x16x4_f32(
            false, Av, false, B, (short)0, cvl, false, false);

        const float lcr = s_lc[li * 3 + 0];
        const float lcg = s_lc[li * 3 + 1];
        const float lcb = s_lc[li * 3 + 2];

#pragma unroll
        for (int r = 0; r < 8; ++r) {
            const float nL = cnl[r];
            const float vL = cvl[r];
            // diffuse term
            const float d = fminf(fmaxf(nL, 0.0f), 1.0f);
            // specular: n.Hv = (n.v + n.L) * rsq(max(||v+L||^2, eps^2))
            const float h2    = vvm[r] + 2.0f * vL + 1.0f;
            const float invhn = __builtin_amdgcn_rsqf(fmaxf(h2, 1e-12f));
            float sc = fminf(fmaxf((nvm[r] + nL) * invhn, 0.0f), 1.0f);
            // pow(sc, shin): log2(0)=-inf -> exp2(-inf)=0; log2(1)=0 -> 1
            const float sp = __builtin_amdgcn_exp2f(shin * __builtin_amdgcn_logf(sc));
            const float w  = kd * d + ksnf * sp;
            accR[r] += lcr * w;
            accG[r] += lcg * w;
            accB[r] += lcb * w;
        }
    }

    // ------- reduce across the 16 light columns (each 16-lane half) --------
#pragma unroll
    for (int off = 1; off < 16; off <<= 1) {
#pragma unroll
        for (int r = 0; r < 8; ++r) {
            accR[r] += __shfl_xor(accR[r], off, 32);
            accG[r] += __shfl_xor(accG[r], off, 32);
            accB[r] += __shfl_xor(accB[r], off, 32);
        }
    }

    // lane 0 owns rows 0-7, lane 16 owns rows 8-15
    if (lane == 0 || lane == 16) {
#pragma unroll
        for (int r = 0; r < 8; ++r) {
            const size_t pp = (size_t)(pixBase + rbase + r) * 3;
            out[pp + 0] = accR[r];
            out[pp + 1] = accG[r];
            out[pp + 2] = accB[r];
        }
    }
}

extern "C" void kernel_launch(void* const* d_in, const int* in_sizes, int n_in,
                              void* d_out, int out_size, void* d_ws, size_t ws_size,
                              hipStream_t stream) {
    (void)in_sizes; (void)n_in; (void)out_size; (void)d_ws; (void)ws_size;
    const float* pn  = (const float*)d_in[0];  // pixel_normals   (H,W,3)
    const float* pd  = (const float*)d_in[1];  // pixel_directions(H,W,3)
    const float* cam = (const float*)d_in[2];  // camera_position (3,)
    const float* ld  = (const float*)d_in[3];  // light_directions(K,3)
    const float* lc  = (const float*)d_in[4];  // light_colors    (K,3)
    const float* sh  = (const float*)d_in[5];  // shininess (1,)
    const float* kd  = (const float*)d_in[6];  // kd (1,)
    const float* ks  = (const float*)d_in[7];  // ks (1,)
    float* out = (float*)d_out;

    // 16 pixels per wave32, 8 waves per block -> 128 pixels per block
    const int blocks = NPIX / 128;  // 2048, exact coverage (512*512)
    blinn_wmma_kernel<<<blocks, 256, 0, stream>>>(pn, pd, cam, ld, lc, sh, kd, ks, out);
}